// PagedOffloadTransformer_54649163874776
// MI455X (gfx1250) — compile-verified
//
#include <hip/hip_runtime.h>

typedef __attribute__((ext_vector_type(2))) float v2f;
typedef __attribute__((ext_vector_type(4))) float v4f;
typedef __attribute__((ext_vector_type(8))) float v8f;

#define DIM   2048
#define HID   8192
#define NH    16
#define HD    128
#define KVBS  16
#define NPART 16
#define PARTT 256
#define ATT_SCALE 0.08838834764831845f   // 1/sqrt(128)
#define EPS 1e-5f

// ---------------------------------------------------------------------------
// fp32 WMMA: D(16x16) += A(16x4) * B(4x16), operands per CDNA5 ISA layout.
// Per lane (half = lane>>4, idx = lane&15):
//   A vgpr pair = A[idx][kb+2*half], A[idx][kb+2*half+1]
//   B vgpr pair = W[n=idx][kb+2*half], W[idx][kb+2*half+1]   (W rows = N cols)
//   D vgpr i    = D[i + 8*half][idx]
// ---------------------------------------------------------------------------
__device__ __forceinline__ v8f wmma4(v2f a, v2f b, v8f c) {
  return __builtin_amdgcn_wmma_f32_16x16x4_f32(false, a, false, b, (short)0, c,
                                               false, false);
}

// Split-K GEMV partial: blockDim.x == 128 (4 waves); wave w covers K/4 of the
// contraction. Weight stream is non-temporal (single pass, don't thrash L2);
// the 16-row padded activation A stays regular-temporal (L2-hot).
// Each wave deposits its 16x16 partial tile into red[wave][m*16+n].
__device__ __forceinline__ void gemv_partial(const float* __restrict__ A, int lda,
                                             const float* __restrict__ W, int ldw,
                                             int K, float red[][256]) {
  const int tid  = threadIdx.x;
  const int wave = tid >> 5;
  const int lane = tid & 31;
  const int half = lane >> 4;
  const int idx  = lane & 15;
  const int kper = K >> 2;
  const int k0   = wave * kper;
  const float* ap = A + (size_t)idx * lda + k0 + 2 * half;
  const float* wp = W + (size_t)idx * ldw + k0 + 2 * half;
  v8f acc = {};
#pragma unroll 8
  for (int kb = 0; kb < kper; kb += 4) {
    v2f a = *(const v2f*)(ap + kb);
    v2f b = __builtin_nontemporal_load((const v2f*)(wp + kb));
    acc = wmma4(a, b, acc);
  }
#pragma unroll
  for (int i = 0; i < 8; ++i)
    red[wave][(i + 8 * half) * 16 + idx] = acc[i];
}

// Sum the 4 wave partials for tile elements e and e+128 (e = tid).
__device__ __forceinline__ v2f tile_reduce(const float red[][256]) {
  const int tid = threadIdx.x;
  v2f r;
  r.x = red[0][tid] + red[1][tid] + red[2][tid] + red[3][tid];
  r.y = red[0][tid + 128] + red[1][tid + 128] + red[2][tid + 128] +
        red[3][tid + 128];
  return r;
}

// ---------------------------------------------------------------------------
// RMSNorm, writes 16 zero-padded rows (rows >= B become 0).
// ---------------------------------------------------------------------------
__global__ void k_rmsnorm(const float* __restrict__ x, const float* __restrict__ g,
                          float* __restrict__ out, int B) {
  const int row = blockIdx.x;
  const int tid = threadIdx.x;
  __shared__ float red[256];
  float ss = 0.f;
  if (row < B)
    for (int i = tid; i < DIM; i += 256) { float v = x[row * DIM + i]; ss += v * v; }
  red[tid] = ss;
  __syncthreads();
  for (int s = 128; s > 0; s >>= 1) {
    if (tid < s) red[tid] += red[tid + s];
    __syncthreads();
  }
  const float inv = rsqrtf(red[0] * (1.0f / DIM) + EPS);
  for (int i = tid; i < DIM; i += 256)
    out[row * DIM + i] = (row < B) ? x[row * DIM + i] * inv * g[i] : 0.f;
}

// ---------------------------------------------------------------------------
// Fused QKV projection: 3 * (2048/16) = 384 output tiles; 4-wave split-K.
// ---------------------------------------------------------------------------
__global__ void __launch_bounds__(128) k_qkv(
    const float* __restrict__ h, const float* __restrict__ wq,
    const float* __restrict__ wk, const float* __restrict__ wv,
    float* __restrict__ q, float* __restrict__ k, float* __restrict__ v) {
  const int t = blockIdx.x;
  const int which = t >> 7;            // DIM/16 == 128 tiles per matrix
  const int e0 = (t & 127) << 4;
  const float* W = (which == 0) ? wq : (which == 1) ? wk : wv;
  float* O       = (which == 0) ? q  : (which == 1) ? k  : v;
  __shared__ float red[4][256];
  gemv_partial(h, DIM, W + (size_t)e0 * DIM, DIM, DIM, red);
  __syncthreads();
  const v2f r = tile_reduce(red);
  const int tid = threadIdx.x;
  O[(size_t)(tid >> 4) * DIM + e0 + (tid & 15)] = r.x;
  O[(size_t)((tid + 128) >> 4) * DIM + e0 + (tid & 15)] = r.y;
}

// ---------------------------------------------------------------------------
// Flash-decode partial: grid (NPART, NH, B), block 128 (4 waves).
// Per token: coalesced non-temporal float4 K/V loads, butterfly dot, online
// softmax. Token l == ctx-1 substitutes the freshly projected k/v row (paged
// cache write without mutating the input heaps).
// ---------------------------------------------------------------------------
__global__ void __launch_bounds__(128) k_attn_part(
    const float* __restrict__ qws, const float* __restrict__ kws,
    const float* __restrict__ vws, const float* __restrict__ kheap,
    const float* __restrict__ vheap, const int* __restrict__ btab,
    const int* __restrict__ ctxl, float* __restrict__ pm,
    float* __restrict__ ps, float* __restrict__ pacc, int MB) {
  const int p = blockIdx.x, hh = blockIdx.y, b = blockIdx.z;
  const int ctx = ctxl[b];
  const int start = p * PARTT;
  const int end = (ctx < start + PARTT) ? ctx : (start + PARTT);
  const int tid = threadIdx.x, wave = tid >> 5, lane = tid & 31;
  const size_t hoff = (size_t)hh * HD + lane * 4;

  const v4f qv = *(const v4f*)(qws + (size_t)b * DIM + hoff);
  float m = -__builtin_inff(), s = 0.f;
  v4f acc = {};

  for (int l = start + wave; l < end; l += 4) {
    const int blk = btab[b * MB + (l >> 4)];
    const size_t tok = ((size_t)blk * KVBS + (l & 15)) * DIM;
    const bool fresh = (l == ctx - 1);
    const v4f kv = fresh
        ? *(const v4f*)(kws + (size_t)b * DIM + hoff)
        : __builtin_nontemporal_load((const v4f*)(kheap + tok + hoff));
    float d = qv.x * kv.x + qv.y * kv.y + qv.z * kv.z + qv.w * kv.w;
#pragma unroll
    for (int off = 16; off > 0; off >>= 1) d += __shfl_xor(d, off, 32);
    const float sc = d * ATT_SCALE;
    const float mn = fmaxf(m, sc);
    const float corr = __expf(m - mn);      // exp(-inf)==0 on first token
    const float w = __expf(sc - mn);
    const v4f vv = fresh
        ? *(const v4f*)(vws + (size_t)b * DIM + hoff)
        : __builtin_nontemporal_load((const v4f*)(vheap + tok + hoff));
    s = s * corr + w;
    acc.x = acc.x * corr + w * vv.x;
    acc.y = acc.y * corr + w * vv.y;
    acc.z = acc.z * corr + w * vv.z;
    acc.w = acc.w * corr + w * vv.w;
    m = mn;
  }

  __shared__ float sm[4], ssum[4];
  __shared__ float sacc[4][HD];
  if (lane == 0) { sm[wave] = m; ssum[wave] = s; }
  sacc[wave][lane * 4 + 0] = acc.x;
  sacc[wave][lane * 4 + 1] = acc.y;
  sacc[wave][lane * 4 + 2] = acc.z;
  sacc[wave][lane * 4 + 3] = acc.w;
  __syncthreads();

  const float gm = fmaxf(fmaxf(sm[0], sm[1]), fmaxf(sm[2], sm[3]));
  float S = 0.f, a = 0.f;
#pragma unroll
  for (int w = 0; w < 4; ++w) {
    if (ssum[w] > 0.f) {                    // guards exp(-inf - -inf)
      const float c = __expf(sm[w] - gm);
      S += ssum[w] * c;
      a += sacc[w][tid] * c;
    }
  }
  const int pidx = (b * NH + hh) * NPART + p;
  if (tid == 0) { pm[pidx] = gm; ps[pidx] = S; }
  pacc[(size_t)pidx * HD + tid] = a;
}

// ---------------------------------------------------------------------------
// Combine NPART partials -> normalized attention output, zero-padded rows.
// grid (NH, 16), block HD.
// ---------------------------------------------------------------------------
__global__ void k_attn_reduce(const float* __restrict__ pm,
                              const float* __restrict__ ps,
                              const float* __restrict__ pacc,
                              float* __restrict__ attn, int B) {
  const int hh = blockIdx.x, row = blockIdx.y, tid = threadIdx.x;
  if (row >= B) { attn[(size_t)row * DIM + hh * HD + tid] = 0.f; return; }
  const int base = (row * NH + hh) * NPART;
  float gm = -__builtin_inff();
  for (int p = 0; p < NPART; ++p) gm = fmaxf(gm, pm[base + p]);
  float S = 0.f, a = 0.f;
  for (int p = 0; p < NPART; ++p) {
    const float sp = ps[base + p];
    if (sp > 0.f) {
      const float c = __expf(pm[base + p] - gm);
      S += sp * c;
      a += pacc[(size_t)(base + p) * HD + tid] * c;
    }
  }
  attn[(size_t)row * DIM + hh * HD + tid] = a / S;
}

// ---------------------------------------------------------------------------
// O projection + residual. grid 128, block 128 (split-K).
// ---------------------------------------------------------------------------
__global__ void __launch_bounds__(128) k_oproj(
    const float* __restrict__ attn, const float* __restrict__ wo,
    const float* __restrict__ x, float* __restrict__ x1, int B) {
  const int e0 = blockIdx.x << 4;
  __shared__ float red[4][256];
  gemv_partial(attn, DIM, wo + (size_t)e0 * DIM, DIM, DIM, red);
  __syncthreads();
  const v2f r = tile_reduce(red);
  const int tid = threadIdx.x;
  {
    const int m = tid >> 4, n = tid & 15;
    const float res = (m < B) ? x[(size_t)m * DIM + e0 + n] : 0.f;
    x1[(size_t)m * DIM + e0 + n] = r.x + res;
  }
  {
    const int m = (tid + 128) >> 4, n = tid & 15;
    const float res = (m < B) ? x[(size_t)m * DIM + e0 + n] : 0.f;
    x1[(size_t)m * DIM + e0 + n] = r.y + res;
  }
}

// ---------------------------------------------------------------------------
// Gate/up projections + SiLU gating: two split-K WMMA tiles (w1, w3) sharing
// the L2-hot activation stream. grid 512, block 128.
// ---------------------------------------------------------------------------
__global__ void __launch_bounds__(128) k_mlp1(
    const float* __restrict__ h2, const float* __restrict__ w1,
    const float* __restrict__ w3, float* __restrict__ gated) {
  const int f0 = blockIdx.x << 4;
  __shared__ float red1[4][256];
  __shared__ float red3[4][256];
  gemv_partial(h2, DIM, w1 + (size_t)f0 * DIM, DIM, DIM, red1);
  gemv_partial(h2, DIM, w3 + (size_t)f0 * DIM, DIM, DIM, red3);
  __syncthreads();
  const v2f r1 = tile_reduce(red1);
  const v2f r3 = tile_reduce(red3);
  const int tid = threadIdx.x;
  const float g0 = r1.x, g1v = r1.y;
  gated[(size_t)(tid >> 4) * HID + f0 + (tid & 15)] =
      (g0 / (1.f + __expf(-g0))) * r3.x;
  gated[(size_t)((tid + 128) >> 4) * HID + f0 + (tid & 15)] =
      (g1v / (1.f + __expf(-g1v))) * r3.y;
}

// ---------------------------------------------------------------------------
// Down projection + residual -> final output. grid 128, block 128 (split-K).
// ---------------------------------------------------------------------------
__global__ void __launch_bounds__(128) k_mlp2(
    const float* __restrict__ gated, const float* __restrict__ w2,
    const float* __restrict__ x1, float* __restrict__ out, int B) {
  const int d0 = blockIdx.x << 4;
  __shared__ float red[4][256];
  gemv_partial(gated, HID, w2 + (size_t)d0 * HID, HID, HID, red);
  __syncthreads();
  const v2f r = tile_reduce(red);
  const int tid = threadIdx.x;
  {
    const int m = tid >> 4, n = tid & 15;
    if (m < B)
      out[(size_t)m * DIM + d0 + n] = r.x + x1[(size_t)m * DIM + d0 + n];
  }
  {
    const int m = (tid + 128) >> 4, n = tid & 15;
    if (m < B)
      out[(size_t)m * DIM + d0 + n] = r.y + x1[(size_t)m * DIM + d0 + n];
  }
}

// ---------------------------------------------------------------------------
extern "C" void kernel_launch(void* const* d_in, const int* in_sizes, int n_in,
                              void* d_out, int out_size, void* d_ws, size_t ws_size,
                              hipStream_t stream) {
  (void)n_in; (void)out_size; (void)ws_size;
  const float* x     = (const float*)d_in[0];
  const float* kheap = (const float*)d_in[1];
  const float* vheap = (const float*)d_in[2];
  const float* wq    = (const float*)d_in[3];
  const float* wk    = (const float*)d_in[4];
  const float* wv    = (const float*)d_in[5];
  const float* wo    = (const float*)d_in[6];
  const float* g1    = (const float*)d_in[7];
  const float* g2    = (const float*)d_in[8];
  const float* w1    = (const float*)d_in[9];
  const float* w2    = (const float*)d_in[10];
  const float* w3    = (const float*)d_in[11];
  const int*   btab  = (const int*)d_in[12];
  const int*   ctxl  = (const int*)d_in[14];

  const int B  = in_sizes[0] / DIM;          // 8
  const int MB = in_sizes[12] / B;           // 257

  // Workspace carve-up (~3.6 MB of fp32; activations padded to 16 rows).
  float* ws      = (float*)d_ws;
  float* h_ws    = ws; ws += 16 * DIM;
  float* q_ws    = ws; ws += 16 * DIM;
  float* k_ws    = ws; ws += 16 * DIM;
  float* v_ws    = ws; ws += 16 * DIM;
  float* attn_ws = ws; ws += 16 * DIM;
  float* x1_ws   = ws; ws += 16 * DIM;
  float* h2_ws   = ws; ws += 16 * DIM;
  float* gat_ws  = ws; ws += 16 * HID;
  float* pm_ws   = ws; ws += 16 * NH * NPART;
  float* ps_ws   = ws; ws += 16 * NH * NPART;
  float* pacc_ws = ws; ws += (size_t)16 * NH * NPART * HD;

  hipLaunchKernelGGL(k_rmsnorm, dim3(16), dim3(256), 0, stream, x, g1, h_ws, B);
  hipLaunchKernelGGL(k_qkv, dim3(384), dim3(128), 0, stream,
                     h_ws, wq, wk, wv, q_ws, k_ws, v_ws);
  hipLaunchKernelGGL(k_attn_part, dim3(NPART, NH, B), dim3(128), 0, stream,
                     q_ws, k_ws, v_ws, kheap, vheap, btab, ctxl,
                     pm_ws, ps_ws, pacc_ws, MB);
  hipLaunchKernelGGL(k_attn_reduce, dim3(NH, 16), dim3(HD), 0, stream,
                     pm_ws, ps_ws, pacc_ws, attn_ws, B);
  hipLaunchKernelGGL(k_oproj, dim3(128), dim3(128), 0, stream,
                     attn_ws, wo, x, x1_ws, B);
  hipLaunchKernelGGL(k_rmsnorm, dim3(16), dim3(256), 0, stream,
                     x1_ws, g2, h2_ws, B);
  hipLaunchKernelGGL(k_mlp1, dim3(512), dim3(128), 0, stream,
                     h2_ws, w1, w3, gat_ws);
  hipLaunchKernelGGL(k_mlp2, dim3(128), dim3(128), 0, stream,
                     gat_ws, w2, x1_ws, (float*)d_out, B);
}